// FusionBlock_3917010174786
// MI455X (gfx1250) — compile-verified
//
#include <hip/hip_runtime.h>

typedef float v2f __attribute__((ext_vector_type(2)));
typedef float v4f __attribute__((ext_vector_type(4)));
typedef float v8f __attribute__((ext_vector_type(8)));

// ---------------------------------------------------------------------------
// GEMM: C[16 x 1024] = A[16 x K] * W^T + bias, where W is [1024 x K] row-major
// (so C[m,n] = sum_k A[m,k] * W[n,k] + bias[n]).
// One wave (32 lanes) per 16-column tile; grid.x = 1024/16 = 64.
// Uses V_WMMA_F32_16X16X4_F32 (fp32-exact, matches the fp32 reference).
// ---------------------------------------------------------------------------
__global__ __launch_bounds__(32) void gemm16_wmma(
    const float* __restrict__ A, int lda,
    const float* __restrict__ W, int ldw,
    const float* __restrict__ bias,
    float* __restrict__ C, int K) {
  const int lane = threadIdx.x;       // 0..31, full wave
  const int half = lane >> 4;         // 0: K={0,1}, 1: K={2,3} per ISA A layout
  const int l16  = lane & 15;
  const int nbase = blockIdx.x << 4;  // output column tile base

  // A-matrix lane pointer: A[m = l16][k0 + 2*half + {0,1}]  (float2, 8B aligned)
  const float* arow = A + (size_t)l16 * lda + 2 * half;
  // B-matrix (= W^T tile) lane pointer: W[nbase + l16][k0 + 2*half + {0,1}]
  const float* wrow = W + (size_t)(nbase + l16) * ldw + 2 * half;

  v8f c = {};
#pragma unroll 4
  for (int k0 = 0; k0 < K; k0 += 4) {
    v2f a = *(const v2f*)(arow + k0);
    v2f b = *(const v2f*)(wrow + k0);
    // 8 args: (neg_a, A, neg_b, B, c_mod, C, reuse_a, reuse_b)
    c = __builtin_amdgcn_wmma_f32_16x16x4_f32(
        false, a, false, b, (short)0, c, false, false);
  }

  // C/D layout: VGPR j -> row (j + 8*half), column nbase + l16
  const int col = nbase + l16;
  const float bval = bias[col];
#pragma unroll
  for (int j = 0; j < 8; ++j) {
    const int row = j + 8 * half;
    C[(size_t)row * 1024 + col] = c[j] + bval;
  }
}

// ---------------------------------------------------------------------------
// delta[i] = out_image[i] + out_audio[i] + out_text[i],  i < 16*1024
// ---------------------------------------------------------------------------
__global__ __launch_bounds__(256) void combine3(
    const float* __restrict__ o, float* __restrict__ delta) {
  const int i = blockIdx.x * 256 + threadIdx.x;
  delta[i] = o[i] + o[16384 + i] + o[32768 + i];
}

// ---------------------------------------------------------------------------
// out[b,l,d] = S[b,l,d] + delta[b,d]  -- 512 MB of streaming traffic, the
// roofline-dominant part. float4 (b128) per lane; non-temporal on the big
// stream so the hot 64 KB delta stays cached while S/out bypass retention.
// Per-batch chunk = L*D/4 = 2^20 float4s; D/4 = 256.
// ---------------------------------------------------------------------------
__global__ __launch_bounds__(256) void bcast_add(
    const float* __restrict__ S, const float* __restrict__ delta,
    float* __restrict__ out) {
  const size_t i = (size_t)blockIdx.x * 256 + threadIdx.x;  // float4 index
  v4f s = __builtin_nontemporal_load((const v4f*)S + i);
  const int d4 = (int)(i & 255);    // D/4 - 1
  const int b  = (int)(i >> 20);    // / (L*D/4)
  v4f dd = *((const v4f*)delta + b * 256 + d4);
  __builtin_nontemporal_store(s + dd, (v4f*)out + i);
}

// ---------------------------------------------------------------------------
// Inputs (dict order):
//  0: S[16,4096,1024]
//  per modality m in {image(din=2048), audio(1024), text(768)} at base 1+7*m:
//    +0 feat[16,din] +1 Wkv[1024,din] +2 bkv[1024]
//    +3 Wv[1024,1024] +4 bv[1024] +5 Wo[1024,1024] +6 bo[1024]
// ---------------------------------------------------------------------------
extern "C" void kernel_launch(void* const* d_in, const int* in_sizes, int n_in,
                              void* d_out, int out_size, void* d_ws, size_t ws_size,
                              hipStream_t stream) {
  (void)in_sizes; (void)n_in; (void)out_size; (void)ws_size;

  const float* S = (const float*)d_in[0];
  float* out = (float*)d_out;

  float* ws    = (float*)d_ws;
  float* kv    = ws;                 // 16*1024
  float* vbuf  = ws + 16 * 1024;     // 16*1024
  float* outm  = ws + 2 * 16 * 1024; // 3 * 16*1024
  float* delta = ws + 5 * 16 * 1024; // 16*1024

  const int dins[3] = {2048, 1024, 768};
  const dim3 gGemm(64), bWave(32);

  for (int m = 0; m < 3; ++m) {
    const int base = 1 + 7 * m;
    const float* feat = (const float*)d_in[base + 0];
    const float* Wkv  = (const float*)d_in[base + 1];
    const float* bkv  = (const float*)d_in[base + 2];
    const float* Wv   = (const float*)d_in[base + 3];
    const float* bv   = (const float*)d_in[base + 4];
    const float* Wo   = (const float*)d_in[base + 5];
    const float* bo   = (const float*)d_in[base + 6];
    const int din = dins[m];

    // kv = feat @ Wkv^T + bkv
    gemm16_wmma<<<gGemm, bWave, 0, stream>>>(feat, din, Wkv, din, bkv, kv, din);
    // v = kv @ Wv^T + bv
    gemm16_wmma<<<gGemm, bWave, 0, stream>>>(kv, 1024, Wv, 1024, bv, vbuf, 1024);
    // out_m = v @ Wo^T + bo
    gemm16_wmma<<<gGemm, bWave, 0, stream>>>(vbuf, 1024, Wo, 1024, bo,
                                             outm + (size_t)m * 16384, 1024);
  }

  // delta = sum of the three modality outputs
  combine3<<<dim3(64), dim3(256), 0, stream>>>(outm, delta);

  // out = S + broadcast(delta): 16*4096*1024 / 4 = 16,777,216 float4s
  bcast_add<<<dim3(65536), dim3(256), 0, stream>>>(S, delta, out);
}